// GCN_9242769621286
// MI455X (gfx1250) — compile-verified
//
#include <hip/hip_runtime.h>
#include <hip/hip_bf16.h>

typedef __attribute__((ext_vector_type(2))) float v2f;
typedef __attribute__((ext_vector_type(8))) float v8f;

#define GCN_D 128
#define BN_EPS 1e-5f

// Force the native global_atomic_add_f32 path (no CAS-loop legalization).
#define GCN_FADD(p, v)                                                        \
  {                                                                           \
    [[clang::atomic(no_remote_memory, no_fine_grained_memory,                 \
                    ignore_denormal_mode)]] {                                 \
      __hip_atomic_fetch_add((p), (v), __ATOMIC_RELAXED,                      \
                             __HIP_MEMORY_SCOPE_AGENT);                       \
    }                                                                         \
  }

// ---------------------------------------------------------------------------
// GEMM: H[N x 128] = A[N x 128] * W[128 x 128] using V_WMMA_F32_16X16X4_F32.
// One wave computes a 16-row x 128-col band: 8 accumulators (v8f), 32 K-steps.
// A fragment (16x4, MxK): lane l -> M = l&15, VGPR r -> K = 2*(l>>4) + r.
// B fragment (4x16, KxN): lane l -> N = l&15, VGPR r -> K = 2*(l>>4) + r.
// C/D (16x16):            lane l -> N = l&15, VGPR r -> M = r + 8*(l>>4).
// ---------------------------------------------------------------------------
__global__ void gcn_gemm_wmma(const float* __restrict__ A,
                              const float* __restrict__ W,
                              float* __restrict__ H,
                              int rowTiles) {
  const int wavesPerBlock = blockDim.x >> 5;
  const int tile = blockIdx.x * wavesPerBlock + (threadIdx.x >> 5);
  if (tile >= rowTiles) return;  // wave-uniform exit; EXEC stays all-ones
  const int lane = threadIdx.x & 31;
  const int m = lane & 15;
  const int half = lane >> 4;            // 0 or 1
  const size_t r0 = (size_t)tile * 16;

  v8f acc[8] = {};
  for (int ks = 0; ks < 32; ++ks) {
    const int c0 = ks * 4;
    // A fragment: float2 load, 8-byte aligned
    v2f a = *(const v2f*)(A + (r0 + m) * GCN_D + c0 + (half << 1));
#pragma unroll
    for (int nt = 0; nt < 8; ++nt) {
      const float* bp = W + (size_t)(c0 + (half << 1)) * GCN_D + nt * 16 + m;
      v2f b;
      b.x = bp[0];
      b.y = bp[GCN_D];
      acc[nt] = __builtin_amdgcn_wmma_f32_16x16x4_f32(
          false, a, false, b, (short)0, acc[nt], false, false);
    }
  }
#pragma unroll
  for (int nt = 0; nt < 8; ++nt) {
#pragma unroll
    for (int r = 0; r < 8; ++r) {
      const size_t row = r0 + r + (half ? 8 : 0);
      H[row * GCN_D + nt * 16 + m] = acc[nt][r];
    }
  }
}

// ---------------------------------------------------------------------------
// Degree = (#edges with dst==n) + 1 (self loop); dinv = rsqrt(deg)
// ---------------------------------------------------------------------------
__global__ void gcn_count_deg(const int* __restrict__ dst, int* __restrict__ deg,
                              int nedges) {
  int i = blockIdx.x * blockDim.x + threadIdx.x;
  if (i < nedges) atomicAdd(&deg[dst[i]], 1);
}

__global__ void gcn_make_dinv(const int* __restrict__ deg, float* __restrict__ dinv,
                              int nnodes) {
  int i = blockIdx.x * blockDim.x + threadIdx.x;
  if (i < nnodes) dinv[i] = rsqrtf((float)(deg[i] + 1));
}

// ---------------------------------------------------------------------------
// agg[n,:] = H[n,:] * dinv[n]^2   (self-loop term; also initializes agg)
// ---------------------------------------------------------------------------
__global__ void gcn_selfloop_init(const float* __restrict__ H,
                                  const float* __restrict__ dinv,
                                  float* __restrict__ agg, int nvec4) {
  int i = blockIdx.x * blockDim.x + threadIdx.x;
  if (i >= nvec4) return;
  int row = i >> 5;  // 128/4 = 32 float4 per row
  float nrm = dinv[row];
  nrm *= nrm;
  float4 v = ((const float4*)H)[i];
  v.x *= nrm; v.y *= nrm; v.z *= nrm; v.w *= nrm;
  ((float4*)agg)[i] = v;
}

// ---------------------------------------------------------------------------
// Edge scatter: one wave32 per edge; lane handles 4 consecutive features.
// agg[dst,:] += H[src,:] * dinv[src]*dinv[dst]  via global_atomic_add_f32.
// Edge id pinned to SGPR so src/dst/dinv loads become scalar loads.
// ---------------------------------------------------------------------------
__global__ void gcn_edge_scatter(const float* __restrict__ H,
                                 const float* __restrict__ dinv,
                                 const int* __restrict__ src,
                                 const int* __restrict__ dst,
                                 float* __restrict__ agg, int nedges) {
  const int wavesPerBlock = blockDim.x >> 5;
  int e = blockIdx.x * wavesPerBlock + (int)(threadIdx.x >> 5);
  e = __builtin_amdgcn_readfirstlane(e);  // wave-uniform -> SGPR
  if (e >= nedges) return;
  const int lane = threadIdx.x & 31;
  const int s = src[e];
  const int d = dst[e];
  const float nrm = dinv[s] * dinv[d];
  float4 v = ((const float4*)(H + (size_t)s * GCN_D))[lane];
  float* out = agg + (size_t)d * GCN_D + lane * 4;
  GCN_FADD(out + 0, v.x * nrm);
  GCN_FADD(out + 1, v.y * nrm);
  GCN_FADD(out + 2, v.z * nrm);
  GCN_FADD(out + 3, v.w * nrm);
}

// ---------------------------------------------------------------------------
// BN stats: per-column sum & sum of squares. 128 threads <-> 128 columns.
// ---------------------------------------------------------------------------
__global__ void gcn_bn_stats(const float* __restrict__ X, float* __restrict__ sums,
                             float* __restrict__ sumsq, int nrows, int rowsPerBlock) {
  const int col = threadIdx.x;
  int r0 = blockIdx.x * rowsPerBlock;
  int r1 = r0 + rowsPerBlock;
  if (r1 > nrows) r1 = nrows;
  float s = 0.f, ss = 0.f;
  for (int r = r0; r < r1; ++r) {
    float v = X[(size_t)r * GCN_D + col];
    s += v;
    ss += v * v;
  }
  GCN_FADD(&sums[col], s);
  GCN_FADD(&sumsq[col], ss);
}

// scale = gamma*rstd ; shift = beta - mu*gamma*rstd
__global__ void gcn_bn_finalize(const float* __restrict__ sums,
                                const float* __restrict__ sumsq,
                                const float* __restrict__ gamma,
                                const float* __restrict__ beta,
                                float* __restrict__ scalev,
                                float* __restrict__ shiftv, int nrows) {
  const int c = threadIdx.x;
  const float invN = 1.0f / (float)nrows;
  float mu = sums[c] * invN;
  float var = sumsq[c] * invN - mu * mu;
  float rstd = rsqrtf(var + BN_EPS);
  float g = gamma[c] * rstd;
  scalev[c] = g;
  shiftv[c] = beta[c] - mu * g;
}

// y = x*scale + shift ; out = y>=0 ? y : a*y   (vectorized float4)
__global__ void gcn_bn_apply_prelu(const float* __restrict__ X,
                                   const float* __restrict__ scalev,
                                   const float* __restrict__ shiftv,
                                   const float* __restrict__ aPtr,
                                   float* __restrict__ out, int nvec4) {
  int i = blockIdx.x * blockDim.x + threadIdx.x;
  if (i >= nvec4) return;
  const int c0 = (i & 31) * 4;  // column of first element in this float4
  const float a = aPtr[0];
  float4 v = ((const float4*)X)[i];
  float y0 = v.x * scalev[c0 + 0] + shiftv[c0 + 0];
  float y1 = v.y * scalev[c0 + 1] + shiftv[c0 + 1];
  float y2 = v.z * scalev[c0 + 2] + shiftv[c0 + 2];
  float y3 = v.w * scalev[c0 + 3] + shiftv[c0 + 3];
  float4 r;
  r.x = (y0 >= 0.f) ? y0 : a * y0;
  r.y = (y1 >= 0.f) ? y1 : a * y1;
  r.z = (y2 >= 0.f) ? y2 : a * y2;
  r.w = (y3 >= 0.f) ? y3 : a * y3;
  ((float4*)out)[i] = r;
}

// ---------------------------------------------------------------------------
extern "C" void kernel_launch(void* const* d_in, const int* in_sizes, int n_in,
                              void* d_out, int out_size, void* d_ws, size_t ws_size,
                              hipStream_t stream) {
  const float* x      = (const float*)d_in[0];
  const int*   eidx   = (const int*)d_in[1];
  const float* W1     = (const float*)d_in[2];
  const float* gamma1 = (const float*)d_in[4];
  const float* beta1  = (const float*)d_in[5];
  const float* a1     = (const float*)d_in[6];
  const float* W2     = (const float*)d_in[7];
  const float* gamma2 = (const float*)d_in[9];
  const float* beta2  = (const float*)d_in[10];
  const float* a2     = (const float*)d_in[11];
  // b1 (d_in[3]) / b2 (d_in[8]) cancel exactly inside BatchNorm -> unused.

  const int N = in_sizes[0] / GCN_D;      // 50000
  const int E = in_sizes[1] / 2;          // 800000
  const int* srcP = eidx;
  const int* dstP = eidx + E;

  // workspace layout
  float* bufA   = (float*)d_ws;                        // N*128 f32
  float* bufB   = bufA + (size_t)N * GCN_D;            // N*128 f32
  int*   deg    = (int*)(bufB + (size_t)N * GCN_D);    // N i32
  float* dinvP  = (float*)(deg + N);                   // N f32
  float* sums   = dinvP + N;                           // 128
  float* sumsq  = sums + GCN_D;                        // 128 (contiguous with sums)
  float* scaleV = sumsq + GCN_D;                       // 128
  float* shiftV = scaleV + GCN_D;                      // 128

  const int rowTiles  = N / 16;                        // 3125 (exact)
  const int gemmBlks  = (rowTiles + 3) / 4;            // 4 waves/block
  const int nvec4     = N * (GCN_D / 4);
  const int evBlks    = (nvec4 + 255) / 256;
  const int edgeBlks  = (E + 7) / 8;                   // 8 waves/block
  const int statBlks  = 200;
  const int rowsPerBlk = (N + statBlks - 1) / statBlks;

  // ---- degrees (shared by both layers) ----
  hipMemsetAsync(deg, 0, (size_t)N * sizeof(int), stream);
  gcn_count_deg<<<(E + 255) / 256, 256, 0, stream>>>(dstP, deg, E);
  gcn_make_dinv<<<(N + 255) / 256, 256, 0, stream>>>(deg, dinvP, N);

  // ---- layer 1 ----
  gcn_gemm_wmma<<<gemmBlks, 128, 0, stream>>>(x, W1, bufA, rowTiles);
  gcn_selfloop_init<<<evBlks, 256, 0, stream>>>(bufA, dinvP, bufB, nvec4);
  gcn_edge_scatter<<<edgeBlks, 256, 0, stream>>>(bufA, dinvP, srcP, dstP, bufB, E);
  hipMemsetAsync(sums, 0, 2 * GCN_D * sizeof(float), stream);
  gcn_bn_stats<<<statBlks, GCN_D, 0, stream>>>(bufB, sums, sumsq, N, rowsPerBlk);
  gcn_bn_finalize<<<1, GCN_D, 0, stream>>>(sums, sumsq, gamma1, beta1, scaleV, shiftV, N);
  gcn_bn_apply_prelu<<<evBlks, 256, 0, stream>>>(bufB, scaleV, shiftV, a1, bufB, nvec4);

  // ---- layer 2 ----
  gcn_gemm_wmma<<<gemmBlks, 128, 0, stream>>>(bufB, W2, bufA, rowTiles);
  gcn_selfloop_init<<<evBlks, 256, 0, stream>>>(bufA, dinvP, bufB, nvec4);
  gcn_edge_scatter<<<edgeBlks, 256, 0, stream>>>(bufA, dinvP, srcP, dstP, bufB, E);
  hipMemsetAsync(sums, 0, 2 * GCN_D * sizeof(float), stream);
  gcn_bn_stats<<<statBlks, GCN_D, 0, stream>>>(bufB, sums, sumsq, N, rowsPerBlk);
  gcn_bn_finalize<<<1, GCN_D, 0, stream>>>(sums, sumsq, gamma2, beta2, scaleV, shiftV, N);
  gcn_bn_apply_prelu<<<evBlks, 256, 0, stream>>>(bufB, scaleV, shiftV, a2, (float*)d_out, nvec4);
}